// GAPnet_seg_2980707303590
// MI455X (gfx1250) — compile-verified
//
#include <hip/hip_runtime.h>
#include <hip/hip_bf16.h>

typedef __attribute__((ext_vector_type(2))) float v2f;
typedef __attribute__((ext_vector_type(8))) float v8f;

#define NPTS 8192
#define KNN 10

// ============================================================================
// Generic fp32 WMMA GEMM: C[M,N] (row-major) = A*B (+bias) (+=C if accum)
// A[m,k] at A[m*a_sm + k*a_sk] ; B[k,n] at B[k*b_sk + n*b_sn]
// M%16==0, N%32==0. One wave per 16x32 tile (two 16x16 accumulators sharing
// the A fragment). K stepped by 4; branch-free main loop, single masked tail.
// Fragment layouts per CDNA5 ISA 7.12.2:
//   A f32 16x4 : lane = (k>>1)*16 + m, vgpr = k&1
//   B f32 4x16 : lane = (k>>1)*16 + n, vgpr = k&1   (mirror of A)
//   C f32 16x16: vgpr r, lanes 0-15 -> M=r,N=lane ; lanes 16-31 -> M=r+8
// ============================================================================
__global__ __launch_bounds__(32) void wmma_gemm_kernel(
    const float* __restrict__ A, long a_sm, long a_sk,
    const float* __restrict__ B, long b_sk, long b_sn,
    const float* __restrict__ bias, float* __restrict__ C,
    int N, int K, int accum)
{
  const int lane = threadIdx.x;
  const int lm = lane & 15;
  const int hi = lane >> 4;
  const long n0 = (long)blockIdx.x * 32;
  const long m0 = (long)blockIdx.y * 16;

  v8f c0, c1;
  if (accum) {
#pragma unroll
    for (int r = 0; r < 8; ++r) {
      c0[r] = C[(m0 + r + 8 * hi) * N + n0 + lm];
      c1[r] = C[(m0 + r + 8 * hi) * N + n0 + 16 + lm];
    }
  } else {
#pragma unroll
    for (int r = 0; r < 8; ++r) { c0[r] = 0.0f; c1[r] = 0.0f; }
  }

  // Base pointers already include this lane's k-phase (2*hi) and m/n position.
  const float* Ap  = A + (m0 + lm) * a_sm + (long)(2 * hi) * a_sk;
  const float* Bp0 = B + (n0 + lm) * b_sn + (long)(2 * hi) * b_sk;
  const float* Bp1 = Bp0 + 16 * b_sn;

  const int Kmain = K & ~3;
  for (int k0 = 0; k0 < Kmain; k0 += 4) {
    const long oa = (long)k0 * a_sk;
    const long ob = (long)k0 * b_sk;
    v2f a, b0, b1;
    a.x  = Ap[oa];
    a.y  = Ap[oa + a_sk];
    b0.x = Bp0[ob];
    b0.y = Bp0[ob + b_sk];
    b1.x = Bp1[ob];
    b1.y = Bp1[ob + b_sk];
    c0 = __builtin_amdgcn_wmma_f32_16x16x4_f32(false, a, false, b0, (short)0, c0,
                                               false, false);
    c1 = __builtin_amdgcn_wmma_f32_16x16x4_f32(false, a, false, b1, (short)0, c1,
                                               false, false);
  }
  if (K & 3) {  // only K=3 and K=67 calls reach here
    const int ka = Kmain + 2 * hi;
    const int kb = ka + 1;
    const long oa = (long)Kmain * a_sk;
    const long ob = (long)Kmain * b_sk;
    v2f a, b0, b1;
    a.x  = (ka < K) ? Ap[oa] : 0.0f;
    a.y  = (kb < K) ? Ap[oa + a_sk] : 0.0f;
    b0.x = (ka < K) ? Bp0[ob] : 0.0f;
    b0.y = (kb < K) ? Bp0[ob + b_sk] : 0.0f;
    b1.x = (ka < K) ? Bp1[ob] : 0.0f;
    b1.y = (kb < K) ? Bp1[ob + b_sk] : 0.0f;
    c0 = __builtin_amdgcn_wmma_f32_16x16x4_f32(false, a, false, b0, (short)0, c0,
                                               false, false);
    c1 = __builtin_amdgcn_wmma_f32_16x16x4_f32(false, a, false, b1, (short)0, c1,
                                               false, false);
  }

#pragma unroll
  for (int r = 0; r < 8; ++r) {
    const long m = m0 + r + 8 * hi;
    float v0 = c0[r];
    float v1 = c1[r];
    if (bias) { v0 += bias[m]; v1 += bias[m]; }
    C[m * N + n0 + lm]      = v0;
    C[m * N + n0 + 16 + lm] = v1;
  }
}

// ============================================================================
// KNN on [8192,3] points (pts[j][d] = x[d*8192+j]); exact top-10 ascending d2
// with lower-index tie-break (matches lax.top_k). Writes gathered [n,10,3].
// ============================================================================
__global__ void knn1_kernel(const float* __restrict__ x, float* __restrict__ xknn1)
{
  int i = blockIdx.x * 256 + threadIdx.x;
  if (i >= NPTS) return;
  float xi = x[i], yi = x[NPTS + i], zi = x[2 * NPTS + i];
  float sqi = xi * xi + yi * yi + zi * zi;
  float dist[KNN];
  int   idx[KNN];
#pragma unroll
  for (int t = 0; t < KNN; ++t) { dist[t] = __builtin_inff(); idx[t] = 0; }
  for (int j = 0; j < NPTS; ++j) {
    float xj = x[j], yj = x[NPTS + j], zj = x[2 * NPTS + j];
    float sqj = xj * xj + yj * yj + zj * zj;
    float dot = xi * xj + yi * yj + zi * zj;
    float d2 = sqi + sqj - 2.0f * dot;
    if (d2 < dist[KNN - 1]) {
      float dv = d2; int iv = j;
#pragma unroll
      for (int s = 0; s < KNN; ++s) {
        if (dv < dist[s]) {
          float td = dist[s]; int ti = idx[s];
          dist[s] = dv; idx[s] = iv; dv = td; iv = ti;
        }
      }
    }
  }
  for (int t = 0; t < KNN; ++t) {
    int j = idx[t];
    xknn1[(long)i * 30 + t * 3 + 0] = x[j];
    xknn1[(long)i * 30 + t * 3 + 1] = x[NPTS + j];
    xknn1[(long)i * 30 + t * 3 + 2] = x[2 * NPTS + j];
  }
}

// Row-wise max of in[M,N] -> out[M]; one block per row.
__global__ void rowmax_kernel(const float* __restrict__ in, float* __restrict__ out, int N)
{
  __shared__ float red[256];
  long row = blockIdx.x;
  int tid = threadIdx.x;
  float m = -__builtin_inff();
  for (int j = tid; j < N; j += 256) m = fmaxf(m, in[row * N + j]);
  red[tid] = m;
  __syncthreads();
  for (int s = 128; s > 0; s >>= 1) {
    if (tid < s) red[tid] = fmaxf(red[tid], red[tid + s]);
    __syncthreads();
  }
  if (tid == 0) out[row] = red[0];
}

// Row-wise sum of squares of in[M,N] -> out[M]; one block per row.
__global__ void rowsumsq_kernel(const float* __restrict__ in, float* __restrict__ out, int N)
{
  __shared__ float red[256];
  long row = blockIdx.x;
  int tid = threadIdx.x;
  float s = 0.0f;
  for (int j = tid; j < N; j += 256) { float v = in[row * N + j]; s += v * v; }
  red[tid] = s;
  __syncthreads();
  for (int q = 128; q > 0; q >>= 1) {
    if (tid < q) red[tid] += red[tid + q];
    __syncthreads();
  }
  if (tid == 0) out[row] = red[0];
}

// y[M] = W[M,K] @ v + b
__global__ void vec_linear_kernel(const float* __restrict__ W, const float* __restrict__ b,
                                  const float* __restrict__ v, float* __restrict__ y,
                                  int M, int K)
{
  int o = blockIdx.x * 256 + threadIdx.x;
  if (o >= M) return;
  float acc = 0.0f;
  for (int k = 0; k < K; ++k) acc += W[(long)o * K + k] * v[k];
  y[o] = acc + b[o];
}

// xnM[3,8192] = (t9.reshape(3,3)+I) @ x   (read later as flat [n,3])
__global__ void xn_kernel(const float* __restrict__ t9, const float* __restrict__ x,
                          float* __restrict__ xnM)
{
  int s = blockIdx.x * 256 + threadIdx.x;
  if (s >= NPTS) return;
  float x0 = x[s], x1 = x[NPTS + s], x2 = x[2 * NPTS + s];
#pragma unroll
  for (int r = 0; r < 3; ++r) {
    float m = (t9[r * 3 + 0] + (r == 0 ? 1.0f : 0.0f)) * x0 +
              (t9[r * 3 + 1] + (r == 1 ? 1.0f : 0.0f)) * x1 +
              (t9[r * 3 + 2] + (r == 2 ? 1.0f : 0.0f)) * x2;
    xnM[(long)r * NPTS + s] = m;
  }
}

// GAP1 pass A: per point, feats (c_flat, 160), logits[n,10], fmax1[n,16]
__global__ void gap1_passA_kernel(const float* __restrict__ xknn1, const float* __restrict__ xnM,
                                  const float* __restrict__ WF, const float* __restrict__ bF,
                                  const float* __restrict__ w1, const float* __restrict__ b1,
                                  float* __restrict__ feats1, float* __restrict__ logits1,
                                  float* __restrict__ fmax1)
{
  int i = blockIdx.x * 128 + threadIdx.x;
  if (i >= NPTS) return;
  float x30[30];
#pragma unroll
  for (int e = 0; e < 30; ++e) x30[e] = xknn1[(long)i * 30 + e];
  // x2[c][t] = x30[c*10+t]  (raw reshape [n,10,3]->[n,3,10])
  float Cf[160];  // c_flat[o*10+t]
#pragma unroll
  for (int o = 0; o < 16; ++o) {
    float w0 = WF[o * 3], wa = WF[o * 3 + 1], wb = WF[o * 3 + 2], bo = bF[o];
#pragma unroll
    for (int t = 0; t < KNN; ++t)
      Cf[o * 10 + t] = w0 * x30[t] + wa * x30[10 + t] + wb * x30[20 + t] + bo;
  }
  // s1 scalar from x1 = xn[i] (flat scramble of [3,8192])
  float xa = xnM[(long)i * 3], xb = xnM[(long)i * 3 + 1], xc = xnM[(long)i * 3 + 2];
  float s1 = 0.0f;
#pragma unroll
  for (int o = 0; o < 16; ++o)
    s1 += w1[o] * (WF[o * 3] * xa + WF[o * 3 + 1] * xb + WF[o * 3 + 2] * xc + bF[o]);
  s1 += b1[0];
#pragma unroll
  for (int t = 0; t < KNN; ++t) {
    float s2 = 0.0f;
#pragma unroll
    for (int o = 0; o < 16; ++o) s2 += w1[o] * Cf[o * 10 + t];
    logits1[(long)i * 10 + t] = s2 + b1[0] + s1;
  }
#pragma unroll
  for (int e = 0; e < 160; ++e) feats1[(long)i * 160 + e] = Cf[e];
  // feats[i][t][f] = Cf[t*16+f] (raw reshape) -> fmax over t
#pragma unroll
  for (int f = 0; f < 16; ++f) {
    float mx = -__builtin_inff();
#pragma unroll
    for (int t = 0; t < KNN; ++t) mx = fmaxf(mx, Cf[t * 16 + f]);
    fmax1[(long)i * 16 + f] = mx;
  }
}

// Column stats for softmax over axis 0: stats[0..9]=colmax, stats[16..25]=sum(exp(x-max))
__global__ void colstats_kernel(const float* __restrict__ logits, float* __restrict__ stats, int n)
{
  __shared__ float red[16][33];
  int tid = threadIdx.x;  // 512
  int col = tid & 15, rg = tid >> 4;  // 32 row-groups
  float mx = -__builtin_inff();
  if (col < KNN)
    for (int r = rg; r < n; r += 32) mx = fmaxf(mx, logits[(long)r * 10 + col]);
  red[col][rg] = mx;
  __syncthreads();
  if (tid < 16) {
    float m = -__builtin_inff();
    for (int j = 0; j < 32; ++j) m = fmaxf(m, red[tid][j]);
    red[tid][32] = m;
    if (tid < KNN) stats[tid] = m;
  }
  __syncthreads();
  float cm = red[col][32];
  float sm = 0.0f;
  if (col < KNN)
    for (int r = rg; r < n; r += 32) sm += expf(logits[(long)r * 10 + col] - cm);
  __syncthreads();
  red[col][rg] = sm;
  __syncthreads();
  if (tid < KNN) {
    float s = 0.0f;
    for (int j = 0; j < 32; ++j) s += red[tid][j];
    stats[16 + tid] = s;
  }
}

// GAP1 pass B: attention-weighted feats + build Hin[67, 8192] for m1 GEMM
__global__ void gap1_passB_kernel(const float* __restrict__ logits1, const float* __restrict__ stats,
                                  const float* __restrict__ feats1, const float* __restrict__ xnM,
                                  float* __restrict__ Hin)
{
  int i = blockIdx.x * 128 + threadIdx.x;
  if (i >= NPTS) return;
  float attn[KNN];
#pragma unroll
  for (int t = 0; t < KNN; ++t)
    attn[t] = expf(logits1[(long)i * 10 + t] - stats[t]) / stats[16 + t];
#pragma unroll
  for (int c = 0; c < 3; ++c) Hin[(long)c * NPTS + i] = xnM[(long)i * 3 + c];
#pragma unroll
  for (int f = 0; f < 16; ++f) {
    float s = 0.0f;
#pragma unroll
    for (int t = 0; t < KNN; ++t) s += attn[t] * feats1[(long)i * 160 + t * 16 + f];
#pragma unroll
    for (int r = 0; r < 4; ++r) Hin[(long)(3 + 4 * f + r) * NPTS + i] = s;  // repeat x4
  }
}

// KNN2 top-10 among 128 "points" of dim 8192 using precomputed Gram + sq.
__global__ void knn2_topk_kernel(const float* __restrict__ G, const float* __restrict__ sq,
                                 int* __restrict__ idx2)
{
  int p = threadIdx.x;
  if (p >= 128) return;
  float dist[KNN];
  int   idx[KNN];
#pragma unroll
  for (int t = 0; t < KNN; ++t) { dist[t] = __builtin_inff(); idx[t] = 0; }
  for (int q = 0; q < 128; ++q) {
    float d2 = sq[p] + sq[q] - 2.0f * G[p * 128 + q];
    if (d2 < dist[KNN - 1]) {
      float dv = d2; int iv = q;
#pragma unroll
      for (int s = 0; s < KNN; ++s) {
        if (dv < dist[s]) {
          float td = dist[s]; int ti = idx[s];
          dist[s] = dv; idx[s] = iv; dv = td; iv = ti;
        }
      }
    }
  }
  for (int t = 0; t < KNN; ++t) idx2[p * 10 + t] = idx[t];
}

// GAP2 pass A: per point (block), WMMA [128x128]x[128x16(pad)] on LDS-staged x2
// gathered on the fly via the double raw-reshape index map.
__global__ __launch_bounds__(256) void gap2_passA_kernel(
    const float* __restrict__ Hm, const int* __restrict__ idx2,
    const float* __restrict__ WF, const float* __restrict__ bF,
    const float* __restrict__ w1, const float* __restrict__ b1,
    float* __restrict__ feats2, float* __restrict__ logits2, float* __restrict__ fmax2)
{
  __shared__ float sX2[128 * 16];
  __shared__ float sC2[128 * 16];
  __shared__ float sX1[128];
  __shared__ float sRed[128];
  __shared__ float sS1;
  int i = blockIdx.x;
  int tid = threadIdx.x;

  // x2[i][c][k] = Hm[idx2[p][t]][d] with g=i*1280+c*10+k, p=i>>6,
  // rem=(i&63)*1280+c*10+k, t=rem>>13, d=rem&8191. Pad k=10..15 with 0.
  for (int e = tid; e < 128 * 16; e += 256) {
    int c = e >> 4, k = e & 15;
    float v = 0.0f;
    if (k < KNN) {
      int p = i >> 6;
      int rem = (i & 63) * 1280 + c * 10 + k;
      int t = rem >> 13;
      int d = rem & 8191;
      int row = idx2[p * 10 + t];
      v = Hm[(long)row * NPTS + d];
    }
    sX2[e] = v;
  }
  if (tid < 128) sX1[tid] = Hm[(long)i * 128 + tid];  // x1 = flat scramble of Hm
  __syncthreads();

  // 8 waves, wave w computes C2 rows [16w,16w+16), K=128 in 32 wmma steps
  int wv = tid >> 5, lane = tid & 31;
  int lm = lane & 15, hi = lane >> 4;
  int m0 = wv * 16;
  v8f acc;
#pragma unroll
  for (int r = 0; r < 8; ++r) acc[r] = 0.0f;
  const float* Ap = WF + (long)(m0 + lm) * 128 + 2 * hi;
#pragma unroll 4
  for (int k0 = 0; k0 < 128; k0 += 4) {
    v2f a, b;
    a.x = Ap[k0];
    a.y = Ap[k0 + 1];
    b.x = sX2[(k0 + 2 * hi) * 16 + lm];
    b.y = sX2[(k0 + 2 * hi + 1) * 16 + lm];
    acc = __builtin_amdgcn_wmma_f32_16x16x4_f32(false, a, false, b, (short)0, acc,
                                                false, false);
  }
#pragma unroll
  for (int r = 0; r < 8; ++r) {
    int m = m0 + r + 8 * hi;
    sC2[m * 16 + lm] = acc[r] + bF[m];
  }
  __syncthreads();

  // feats2 store: c_flat[o*10+t], 1280 values
  for (int e = tid; e < 1280; e += 256)
    feats2[(long)i * 1280 + e] = sC2[(e / 10) * 16 + (e % 10)];

  // s1 = w1 . (WF@x1 + bF) + b1
  float contrib = 0.0f;
  if (tid < 128) {
    float y = 0.0f;
    for (int c = 0; c < 128; ++c) y += WF[(long)tid * 128 + c] * sX1[c];
    contrib = w1[tid] * (y + bF[tid]);
  }
  if (tid < 128) sRed[tid] = contrib;
  __syncthreads();
  for (int s = 64; s > 0; s >>= 1) {
    if (tid < s) sRed[tid] += sRed[tid + s];
    __syncthreads();
  }
  if (tid == 0) sS1 = sRed[0] + b1[0];
  __syncthreads();

  if (tid < KNN) {
    float s2 = 0.0f;
    for (int o = 0; o < 128; ++o) s2 += w1[o] * sC2[o * 16 + tid];
    logits2[(long)i * 10 + tid] = s2 + b1[0] + sS1;
  }
  // fmax over k of feats[i][k][f] = c_flat[k*128+f]
  if (tid < 128) {
    float mx = -__builtin_inff();
#pragma unroll
    for (int k = 0; k < KNN; ++k) {
      int j = k * 128 + tid;
      mx = fmaxf(mx, sC2[(j / 10) * 16 + (j % 10)]);
    }
    fmax2[(long)i * 128 + tid] = mx;
  }
}

// GAP2 pass B: x2a[n,128] = attn . feats
__global__ void gap2_passB_kernel(const float* __restrict__ logits2, const float* __restrict__ stats,
                                  const float* __restrict__ feats2, float* __restrict__ x2a)
{
  __shared__ float attn[KNN];
  int i = blockIdx.x, tid = threadIdx.x;  // 128 threads
  if (tid < KNN)
    attn[tid] = expf(logits2[(long)i * 10 + tid] - stats[tid]) / stats[16 + tid];
  __syncthreads();
  float s = 0.0f;
#pragma unroll
  for (int t = 0; t < KNN; ++t) s += attn[t] * feats2[(long)i * 1280 + t * 128 + tid];
  x2a[(long)i * 128 + tid] = s;
}

// Fold 4 adjacent columns (jnp.repeat(...,4) absorbed into weights):
// out[o*F+f] = sum_{r<4} W[o*ldw + coloff + 4f + r]
__global__ void fold4_kernel(const float* __restrict__ W, int ldw, int coloff,
                             float* __restrict__ out, int total, int F)
{
  int e = blockIdx.x * 256 + threadIdx.x;
  if (e >= total) return;
  int o = e / F, f = e % F;
  const float* wp = W + (long)o * ldw + coloff + 4 * f;
  out[e] = wp[0] + wp[1] + wp[2] + wp[3];
}

// ============================================================================
// Host orchestration
// ============================================================================
static inline void gemm(hipStream_t st, const float* A, long a_sm, long a_sk,
                        const float* B, long b_sk, long b_sn, const float* bias,
                        float* C, int M, int N, int K, int accum)
{
  dim3 grid(N / 32, M / 16);
  wmma_gemm_kernel<<<grid, 32, 0, st>>>(A, a_sm, a_sk, B, b_sk, b_sn, bias, C, N, K, accum);
}

// Workspace layout (float offsets); peak usage ~94 MB.
enum : long {
  O_XKNN1 = 0,                       // 8192*30
  O_XNM   = 245760,                  // 3*8192
  O_FMAX1 = 270336,                  // 8192*16
  O_FMAX2 = 401408,                  // 8192*128
  O_LOG   = 1449984,                 // 8192*10 (reused gap1/gap2)
  O_STATS = 1531904,                 // 32
  O_FEATS1= 1531936,                 // 8192*160
  O_HIN   = 2842656,                 // 67*8192
  O_H1    = 3391520,                 // 64*8192
  O_H2    = 3915808,                 // 64*8192
  O_HM    = 4440096,                 // 128*8192
  O_GMAT  = 5488672,                 // 128*128
  O_SQ2   = 5505056,                 // 128
  O_IDX2  = 5505184,                 // 128*10 (int)
  O_TV    = 5506464,                 // 1024
  O_TL1   = 5507488,                 // 512
  O_TL2   = 5508000,                 // 256
  O_T9    = 5508256,                 // 16
  O_POOL  = 5508272,                 // 1024
  O_V1    = 5509296,                 // 2048
  O_V2    = 5511344,                 // 1024
  O_V3    = 5512368,                 // 512
  O_V4    = 5512880,                 // 256
  O_M4F   = 5513136,                 // 128*128
  O_M7G1  = 5529520,                 // 1024*16
  O_M7G2  = 5545904,                 // 1024*128
  O_X2A   = 5676976,                 // 8192*128
  O_A1    = 6725552,                 // 128*8192
  O_A2    = 7774128,                 // 128*8192
  O_A3    = 8822704,                 // 512*8192
  O_BIG   = 13017008,                // 10485760 (reused: tnet T1/T2/T3, feats2, M7)
};

extern "C" void kernel_launch(void* const* d_in, const int* in_sizes, int n_in,
                              void* d_out, int out_size, void* d_ws, size_t ws_size,
                              hipStream_t stream)
{
  (void)in_sizes; (void)n_in; (void)out_size; (void)ws_size;
  const float* x = (const float*)d_in[0];
  // params: W at 1+2k, b at 2+2k (dict insertion order)
  const float *Wtm1 = (const float*)d_in[1],  *btm1 = (const float*)d_in[2];
  const float *Wtm2 = (const float*)d_in[3],  *btm2 = (const float*)d_in[4];
  const float *Wtm3 = (const float*)d_in[5],  *btm3 = (const float*)d_in[6];
  const float *Wtf1 = (const float*)d_in[7],  *btf1 = (const float*)d_in[8];
  const float *Wtf2 = (const float*)d_in[9],  *btf2 = (const float*)d_in[10];
  const float *Wtf3 = (const float*)d_in[11], *btf3 = (const float*)d_in[12];
  const float *Wg1F = (const float*)d_in[13], *bg1F = (const float*)d_in[14];
  const float *Wg11 = (const float*)d_in[15], *bg11 = (const float*)d_in[16];
  const float *Wg2F = (const float*)d_in[17], *bg2F = (const float*)d_in[18];
  const float *Wg21 = (const float*)d_in[19], *bg21 = (const float*)d_in[20];
  const float *Wm1  = (const float*)d_in[21], *bm1  = (const float*)d_in[22];
  const float *Wm2  = (const float*)d_in[23], *bm2  = (const float*)d_in[24];
  const float *Wm3  = (const float*)d_in[25], *bm3  = (const float*)d_in[26];
  const float *Wm4  = (const float*)d_in[27], *bm4  = (const float*)d_in[28];
  const float *Wm5  = (const float*)d_in[29], *bm5  = (const float*)d_in[30];
  const float *Wm6  = (const float*)d_in[31], *bm6  = (const float*)d_in[32];
  const float *Wm7  = (const float*)d_in[33], *bm7  = (const float*)d_in[34];
  const float *Wf1  = (const float*)d_in[35], *bf1  = (const float*)d_in[36];
  const float *Wf2  = (const float*)d_in[37], *bf2  = (const float*)d_in[38];
  const float *Wf3  = (const float*)d_in[39], *bf3  = (const float*)d_in[40];
  const float *Wf4  = (const float*)d_in[41], *bf4  = (const float*)d_in[42];
  const float *Wf5  = (const float*)d_in[43], *bf5  = (const float*)d_in[44];

  float* w = (float*)d_ws;
  float* T1 = w + O_BIG;                // 64*8192
  float* T2 = w + O_BIG + 524288;       // 128*8192
  float* T3 = w + O_BIG + 1572864;      // 1024*8192
  float* FEATS2 = w + O_BIG;            // 8192*1280 (after tnet done)
  float* M7 = w + O_BIG;                // 1024*8192 (after gap2 done)
  int* idx2 = (int*)(w + O_IDX2);

  // --- KNN1 ---
  knn1_kernel<<<NPTS / 256, 256, 0, stream>>>(x, w + O_XKNN1);

  // --- TNet ---
  gemm(stream, Wtm1, 3, 1,    x,       NPTS, 1, btm1, T1, 64,   NPTS, 3,   0);
  gemm(stream, Wtm2, 64, 1,   T1,      NPTS, 1, btm2, T2, 128,  NPTS, 64,  0);
  gemm(stream, Wtm3, 128, 1,  T2,      NPTS, 1, btm3, T3, 1024, NPTS, 128, 0);
  rowmax_kernel<<<1024, 256, 0, stream>>>(T3, w + O_TV, NPTS);
  vec_linear_kernel<<<8, 256, 0, stream>>>(Wtf1, btf1, w + O_TV,  w + O_TL1, 512, 1024);
  vec_linear_kernel<<<1, 256, 0, stream>>>(Wtf2, btf2, w + O_TL1, w + O_TL2, 256, 512);
  vec_linear_kernel<<<1, 256, 0, stream>>>(Wtf3, btf3, w + O_TL2, w + O_T9,  9,   256);
  xn_kernel<<<NPTS / 256, 256, 0, stream>>>(w + O_T9, x, w + O_XNM);

  // --- GAP1 ---
  gap1_passA_kernel<<<NPTS / 128, 128, 0, stream>>>(w + O_XKNN1, w + O_XNM, Wg1F, bg1F,
                                                    Wg11, bg11, w + O_FEATS1, w + O_LOG,
                                                    w + O_FMAX1);
  colstats_kernel<<<1, 512, 0, stream>>>(w + O_LOG, w + O_STATS, NPTS);
  gap1_passB_kernel<<<NPTS / 128, 128, 0, stream>>>(w + O_LOG, w + O_STATS, w + O_FEATS1,
                                                    w + O_XNM, w + O_HIN);

  // --- h chain (m1,m2,m3): Hm[128,8192] == h.T ---
  gemm(stream, Wm1, 67, 1, w + O_HIN, NPTS, 1, bm1, w + O_H1, 64,  NPTS, 67, 0);
  gemm(stream, Wm2, 64, 1, w + O_H1,  NPTS, 1, bm2, w + O_H2, 64,  NPTS, 64, 0);
  gemm(stream, Wm3, 64, 1, w + O_H2,  NPTS, 1, bm3, w + O_HM, 128, NPTS, 64, 0);

  // --- KNN2 on 128 points of dim 8192: G = Hm @ Hm^T ---
  gemm(stream, w + O_HM, NPTS, 1, w + O_HM, 1, NPTS, nullptr, w + O_GMAT, 128, 128, NPTS, 0);
  rowsumsq_kernel<<<128, 256, 0, stream>>>(w + O_HM, w + O_SQ2, NPTS);
  knn2_topk_kernel<<<1, 128, 0, stream>>>(w + O_GMAT, w + O_SQ2, idx2);

  // --- GAP2 ---
  gap2_passA_kernel<<<NPTS, 256, 0, stream>>>(w + O_HM, idx2, Wg2F, bg2F, Wg21, bg21,
                                              FEATS2, w + O_LOG, w + O_FMAX2);
  colstats_kernel<<<1, 512, 0, stream>>>(w + O_LOG, w + O_STATS, NPTS);
  gap2_passB_kernel<<<NPTS, 128, 0, stream>>>(w + O_LOG, w + O_STATS, FEATS2, w + O_X2A);

  // --- a chain (m4 folded for repeat4, m5, m6) ---
  fold4_kernel<<<(128 * 128 + 255) / 256, 256, 0, stream>>>(Wm4, 512, 0, w + O_M4F,
                                                            128 * 128, 128);
  gemm(stream, w + O_M4F, 128, 1, w + O_X2A, 1, 128, bm4, w + O_A1, 128, NPTS, 128, 0);
  gemm(stream, Wm5, 128, 1, w + O_A1, NPTS, 1, bm5, w + O_A2, 128, NPTS, 128, 0);
  gemm(stream, Wm6, 128, 1, w + O_A2, NPTS, 1, bm6, w + O_A3, 512, NPTS, 128, 0);

  // --- m7 on concat([xg1, xg2, a]) as three accumulating GEMMs ---
  fold4_kernel<<<(1024 * 16 + 255) / 256, 256, 0, stream>>>(Wm7, 1088, 0, w + O_M7G1,
                                                            1024 * 16, 16);
  fold4_kernel<<<(1024 * 128 + 255) / 256, 256, 0, stream>>>(Wm7, 1088, 64, w + O_M7G2,
                                                             1024 * 128, 128);
  gemm(stream, w + O_M7G1, 16, 1,  w + O_FMAX1, 1, 16,   bm7,     M7, 1024, NPTS, 16,  0);
  gemm(stream, w + O_M7G2, 128, 1, w + O_FMAX2, 1, 128,  nullptr, M7, 1024, NPTS, 128, 1);
  gemm(stream, Wm7 + 576, 1088, 1, w + O_A3,    NPTS, 1, nullptr, M7, 1024, NPTS, 512, 1);
  rowmax_kernel<<<1024, 256, 0, stream>>>(M7, w + O_POOL, NPTS);

  // --- classifier MLP ---
  vec_linear_kernel<<<8, 256, 0, stream>>>(Wf1, bf1, w + O_POOL, w + O_V1, 2048, 1024);
  vec_linear_kernel<<<4, 256, 0, stream>>>(Wf2, bf2, w + O_V1,   w + O_V2, 1024, 2048);
  vec_linear_kernel<<<2, 256, 0, stream>>>(Wf3, bf3, w + O_V2,   w + O_V3, 512,  1024);
  vec_linear_kernel<<<1, 256, 0, stream>>>(Wf4, bf4, w + O_V3,   w + O_V4, 256,  512);
  vec_linear_kernel<<<1, 256, 0, stream>>>(Wf5, bf5, w + O_V4,   (float*)d_out, 3, 256);
}